// LIF_neurons_comb_time_MAP_SNN_82506321756602
// MI455X (gfx1250) — compile-verified
//
#include <hip/hip_runtime.h>
#include <stdint.h>

// ---------------------------------------------------------------------------
// LIF spiking-neuron scan for MI455X (gfx1250).
// One thread owns one (b, n) column; recurrence state lives in a register.
// Input streamed through LDS via CDNA5 async global->LDS copies (ASYNCcnt),
// triple-buffered in 16-timestep chunks so ~2 chunks (4 KB/wave, ~16 MB
// device-wide across 4096 waves) stay in flight -> saturates 23.3 TB/s HBM.
// ---------------------------------------------------------------------------

#define AS1 __attribute__((address_space(1)))
#define AS3 __attribute__((address_space(3)))

#define BLOCK 256
#define CHUNK 16         // timesteps per prefetch chunk
#define NBUF  3          // triple buffer; max outstanding = 3*CHUNK = 48 <= 63

// ---- CDNA5 async copy: global -> LDS (tracked by ASYNCcnt) ----------------
#if defined(__has_builtin) && __has_builtin(__builtin_amdgcn_global_load_async_to_lds_b32)
__device__ __forceinline__ void async_copy_b32(const float* gsrc, AS3 float* ldst) {
  __builtin_amdgcn_global_load_async_to_lds_b32(
      (AS1 int*)const_cast<float*>(gsrc), (AS3 int*)ldst,
      /*imm offset*/0, /*cpol*/0);
}
#else
__device__ __forceinline__ void async_copy_b32(const float* gsrc, AS3 float* ldst) {
  unsigned lds_off = (unsigned)(uintptr_t)ldst;
  asm volatile("global_load_async_to_lds_b32 %0, %1, off"
               :: "v"(lds_off), "v"(gsrc) : "memory");
}
#endif

// ---- s_wait_asynccnt <= n (n must be a compile-time constant) -------------
#if defined(__has_builtin) && __has_builtin(__builtin_amdgcn_s_wait_asynccnt)
#define WAIT_ASYNC(n) do { __builtin_amdgcn_s_wait_asynccnt(n); \
                           asm volatile("" ::: "memory"); } while (0)
#else
#define WAIT_ASYNC(n) asm volatile("s_wait_asynccnt %0" :: "n"(n) : "memory")
#endif

__global__ __launch_bounds__(BLOCK)
void lif_scan_kernel(const float* __restrict__ x,
                     const float* __restrict__ h_decay,
                     const float* __restrict__ h_thresh,
                     const float* __restrict__ h_inh,
                     float* __restrict__ out,
                     int T, int N, int nmask, int bshift)
{
  __shared__ float smem[NBUF][CHUNK][BLOCK];   // 48 KB

  const int tid = threadIdx.x;
  const int gid = blockIdx.x * BLOCK + tid;    // flat (b, n); BLOCK divides N
  const int n   = gid & nmask;                 // N is a power of two
  const int b   = gid >> bshift;

  // ---- per-neuron constants (loaded once) ----
  const float hd = h_decay[n];
  const float th = h_thresh[n];
  const float hi = h_inh[n];

  const float decay        = fminf(fmaxf(hd * 6.0f, 0.0f), 6.0f) * (1.0f / 6.0f);
  const float inh_m1       = hi - 1.0f;
  const float log2_inh     = __builtin_amdgcn_logf(hi);   // v_log_f32 (base-2)
  const float inv_log2_inh = 1.0f / log2_inh;             // log(y)/log(inh) == log2(y)/log2(inh)
  const float c1           = inh_m1 / th;
  const float c2           = th / inh_m1;

  const size_t base = (size_t)b * (size_t)T * (size_t)N + (size_t)n;
  const float* gx = x   + base;
  float*       gy = out + base;

  float mem = 0.0f;
  const int nchunks = T / CHUNK;               // T=1024, CHUNK=16 -> 64

  // ---- prime the pipeline: chunks 0..NBUF-1 (48 asyncs outstanding) ----
  #pragma unroll
  for (int c = 0; c < NBUF; ++c) {
    const float* src = gx + (size_t)c * CHUNK * N;
    #pragma unroll
    for (int i = 0; i < CHUNK; ++i)
      async_copy_b32(src + (size_t)i * N, (AS3 float*)&smem[c][i][tid]);
  }

  int buf = 0;
  for (int k = 0; k < nchunks; ++k) {
    // chunks complete in issue order; wait until everything older than the
    // chunks still legitimately in flight has landed -> chunk k is ready.
    if (k < nchunks - 2)       WAIT_ASYNC(2 * CHUNK);
    else if (k == nchunks - 2) WAIT_ASYNC(CHUNK);
    else                       WAIT_ASYNC(0);

    // ---- consume chunk k from smem[buf] ----
    float* dst = gy + (size_t)k * CHUNK * N;
    #pragma unroll
    for (int i = 0; i < CHUNK; ++i) {
      const float xv       = smem[buf][i][tid];           // ds_load (2addr-fused)
      const float mem_temp = mem + xv;
      const float r        = fmaxf(mem_temp, 0.0f);
      const float n_rf     = __builtin_amdgcn_logf(fmaf(r, c1, 1.0f)) * inv_log2_inh;
      const float spike    = (mem_temp > 0.0f) ? floorf(n_rf) : 0.0f;
      const float mem_rst  = (__builtin_amdgcn_exp2f(spike * log2_inh) - 1.0f) * c2;
      mem = (mem_temp - mem_rst) * decay;
      __builtin_nontemporal_store(spike, dst + (size_t)i * N);  // streamed output
    }

    // ---- refill this buffer with chunk k+NBUF (safe: chunk k consumed) ----
    if (k + NBUF < nchunks) {
      const float* src = gx + (size_t)(k + NBUF) * CHUNK * N;
      #pragma unroll
      for (int i = 0; i < CHUNK; ++i)
        async_copy_b32(src + (size_t)i * N, (AS3 float*)&smem[buf][i][tid]);
    }

    buf = (buf + 1 == NBUF) ? 0 : buf + 1;
  }
}

extern "C" void kernel_launch(void* const* d_in, const int* in_sizes, int n_in,
                              void* d_out, int out_size, void* d_ws, size_t ws_size,
                              hipStream_t stream) {
  const float* x  = (const float*)d_in[0];   // (B, T, N) fp32
  const float* hd = (const float*)d_in[1];   // (N,)
  const float* ht = (const float*)d_in[2];   // (N,)
  const float* hi = (const float*)d_in[3];   // (N,)
  float* out = (float*)d_out;                // (B, T, N) fp32

  const int N = in_sizes[1];                 // 2048 (power of two)
  const int T = 1024;                        // reference fixes T
  const long long total = (long long)in_sizes[0];
  const int B = (int)(total / ((long long)T * (long long)N));   // 64

  int bshift = 0;
  while ((1 << bshift) < N) ++bshift;        // log2(N)
  const int nmask = N - 1;

  const int threads = B * N;                 // 131072 -> 4096 wave32
  const int grid    = threads / BLOCK;       // 512 blocks

  lif_scan_kernel<<<grid, BLOCK, 0, stream>>>(x, hd, ht, hi, out, T, N, nmask, bshift);
}